// HiPPO_4355096838801
// MI455X (gfx1250) — compile-verified
//
#include <hip/hip_runtime.h>

#define N      128
#define SEQL   2048
#define NBATCH 2

typedef float v2f __attribute__((ext_vector_type(2)));
typedef float v4f __attribute__((ext_vector_type(4)));
typedef float v8f __attribute__((ext_vector_type(8)));
typedef unsigned int  v4u __attribute__((ext_vector_type(4)));
typedef int           v4i __attribute__((ext_vector_type(4)));
typedef int           v8i __attribute__((ext_vector_type(8)));

// ---------------------------------------------------------------------------
// Kernel 1: bilinear (GBT alpha=0.5) discretization by forward substitution.
// A is lower triangular (HiPPO-LegS), p1 = I - 0.5A.  Thread `col` owns one
// RHS column of [I + 0.5A]; Bd is done by thread 0 afterwards.
// ---------------------------------------------------------------------------
__global__ __launch_bounds__(128) void k_solve(const float* __restrict__ A,
                                               const float* __restrict__ Bv,
                                               float* __restrict__ Ad,
                                               float* __restrict__ Bd) {
  __shared__ float X[N * N];
  const int col = threadIdx.x;
  for (int i = 0; i < N; ++i) {
    float s = ((i == col) ? 1.0f : 0.0f) + 0.5f * A[i * N + col];
    for (int j = 0; j < i; ++j)
      s += 0.5f * A[i * N + j] * X[j * N + col];   // -p1[i][j] = +0.5*A[i][j]
    s /= (1.0f - 0.5f * A[i * N + i]);
    X[i * N + col] = s;
    Ad[i * N + col] = s;
  }
  __syncthreads();
  if (col == 0) {
    for (int i = 0; i < N; ++i) {
      float s = Bv[i];
      for (int j = 0; j < i; ++j)
        s += 0.5f * A[i * N + j] * X[j];
      s /= (1.0f - 0.5f * A[i * N + i]);
      X[i] = s;
      Bd[i] = s;
    }
  }
}

// ---------------------------------------------------------------------------
// Kernel 2: init.  Kext rows [0,128) = 0 (causal pad), row 128 = K[0] = Bd,
// M0 = Ad (first power for doubling).
// ---------------------------------------------------------------------------
__global__ __launch_bounds__(256) void k_init(float* __restrict__ Kext,
                                              const float* __restrict__ Ad,
                                              const float* __restrict__ Bd,
                                              float* __restrict__ M0) {
  int idx = blockIdx.x * 256 + threadIdx.x;
  if (idx < N * N) { Kext[idx] = 0.0f; M0[idx] = Ad[idx]; }
  if (idx < N)     Kext[N * N + idx] = Bd[idx];
}

// ---------------------------------------------------------------------------
// Kernel 3: impulse-response doubling.  K[len+i] = M @ K[i], M = Ad^len.
// ---------------------------------------------------------------------------
__global__ __launch_bounds__(256) void k_expand(float* __restrict__ Kext,
                                                const float* __restrict__ M,
                                                int len) {
  int idx = blockIdx.x * 256 + threadIdx.x;
  if (idx >= len * N) return;
  int i = idx >> 7, n = idx & (N - 1);
  const float* krow = Kext + (size_t)(N + i) * N;
  const float* mrow = M + (size_t)n * N;
  float s = 0.0f;
  for (int m = 0; m < N; ++m) s += krow[m] * mrow[m];
  Kext[(size_t)(N + len + i) * N + n] = s;
}

// Kernel 4: matrix squaring  Mout = Min @ Min.
__global__ __launch_bounds__(256) void k_square(const float* __restrict__ Min,
                                                float* __restrict__ Mout) {
  int idx = blockIdx.x * 256 + threadIdx.x;
  int n = idx >> 7, m = idx & (N - 1);
  float s = 0.0f;
  for (int p = 0; p < N; ++p) s += Min[n * N + p] * Min[p * N + m];
  Mout[n * N + m] = s;
}

// ---------------------------------------------------------------------------
// TDM descriptor issue: DMA one 256x128 fp32 K-segment (row-major, 128-DWORD
// rows) from global into LDS at `lds_off`, padding +4 DWORDs after every 128
// DWORDs -> LDS pitch of 132 floats (bank-conflict avoidance), tracked by
// TENSORcnt.  D# layout per CDNA5 ISA ch.8 (groups 0/1; 2D tile -> g2=g3=0).
// This toolchain exposes the 6-arg builtin: (g0, g1, g2, g3, g4, cpol).
// ---------------------------------------------------------------------------
__device__ __forceinline__ void tdm_load_kseg(const float* gsrc, unsigned lds_off) {
  unsigned long long ga = (unsigned long long)(size_t)gsrc;
  v4u g0;
  g0[0] = 1u;                                             // count=1, user mode
  g0[1] = lds_off;                                        // lds_addr (bytes)
  g0[2] = (unsigned)(ga & 0xFFFFFFFFu);                   // global_addr[31:0]
  g0[3] = (unsigned)((ga >> 32) & 0x1FFFFFFu) | (2u << 30); // addr[56:32]|type=2
  v8i g1;
  g1[0] = (int)((2u << 16)      // data_size = 4B
              | (1u << 20)      // pad_enable
              | (6u << 22)      // pad_interval: 128 DWORDs
              | (3u << 25));    // pad_amount:   4 DWORDs
  g1[1] = (int)(128u  << 16);   // tensor_dim0[15:0] in bits[63:48]
  g1[2] = (int)(2176u << 16);   // tensor_dim0 hi=0, tensor_dim1[15:0]
  g1[3] = (int)(128u  << 16);   // tensor_dim1 hi=0, tile_dim0=128
  g1[4] = 256;                  // tile_dim1=256, tile_dim2=0
  g1[5] = 128;                  // tensor_dim0_stride[31:0]
  g1[6] = 0;                    // stride hi, tensor_dim1_stride lo (unused, 2D)
  g1[7] = 0;
  v4i z4 = {0, 0, 0, 0};
  v8i z8 = {0, 0, 0, 0, 0, 0, 0, 0};
  __builtin_amdgcn_tensor_load_to_lds(g0, g1, z4, z4, z8, 0);
}

// ---------------------------------------------------------------------------
// Kernel 5: causal convolution as banded-Toeplitz GEMM with fp32 WMMA.
//   c[b, it*128+tj, n] = sum_is sum_u  Tm[tj,u] * Kext[(it-is)*128 + u, n]
//   Tm[tj,u] = f[b, is*128 + 128+tj-u]  for u in [tj+1, tj+128], else 0
// One workgroup per (it, b); 8 waves x 8 n-tiles of 16x16, K stepped by 4.
// K-segments are double-buffered in LDS and fetched by the Tensor Data Mover
// (wave 0 issues, TENSORcnt-pipelined) while all waves run WMMA on the other
// buffer.  The f-window is stored REVERSED so the Toeplitz A-operand is an
// ascending contiguous LDS pair (no swap movs).
// ---------------------------------------------------------------------------
#define KT_P 132   // LDS pitch (floats) produced by the TDM pad feature
__global__ __launch_bounds__(256) void k_conv(const float* __restrict__ f,
                                              const float* __restrict__ Kext,
                                              float* __restrict__ cbuf) {
  __shared__ float Kt[2][256 * KT_P];   // 2 x 135168 B (<= 320 KB/WGP)
  __shared__ float frev[384];           // reversed, zero-padded f window
  const int it  = blockIdx.x;
  const int b   = blockIdx.y;
  const int tid = threadIdx.x;
  const int lane = tid & 31;
  const int w    = tid >> 5;          // wave id 0..7 -> output rows [16w,16w+16)
  const int hi   = lane >> 4;         // lane half selects K pair
  const int l15  = lane & 15;
  const int tj   = (w << 4) + l15;    // A-operand row
  const bool wave0 = (__builtin_amdgcn_readfirstlane(tid) < 32);

  v8f acc[8];
#pragma unroll
  for (int nt = 0; nt < 8; ++nt) {
    v8f z = {0.f, 0.f, 0.f, 0.f, 0.f, 0.f, 0.f, 0.f};
    acc[nt] = z;
  }
  if (tid < 128) { frev[tid] = 0.0f; frev[256 + tid] = 0.0f; }

  // prologue: kick off DMA of segment for is=0 into buffer 0
  if (wave0)
    tdm_load_kseg(Kext + (size_t)it * 128 * N,
                  (unsigned)(size_t)&Kt[0][0]);

  for (int is = 0; is <= it; ++is) {
    const int cur = is & 1;
    __syncthreads();   // all waves finished reading Kt[cur^1] (prev iteration)
    if (wave0) {
      if (is < it)     // prefetch next segment into the other buffer
        tdm_load_kseg(Kext + (size_t)(it - is - 1) * 128 * N,
                      (unsigned)(size_t)&Kt[cur ^ 1][0]);
      if (is < it) __builtin_amdgcn_s_wait_tensorcnt(1);  // Kt[cur] ready
      else         __builtin_amdgcn_s_wait_tensorcnt(0);
    }
    // reversed f window: frev[128+x] = f_blk[127-x]
    if (tid < 128) frev[128 + tid] = f[b * SEQL + is * 128 + 127 - tid];
    __syncthreads();   // Kt[cur] + frev visible to all waves

#pragma unroll 2
    for (int u0 = 0; u0 < 256; u0 += 4) {
      // A (16x4 Toeplitz of f): lanes0-15 K={u0,u0+1}, lanes16-31 K={u0+2,u0+3}
      const int jb = 127 - tj + u0 + (hi << 1);
      v2f a;
      a.x = frev[jb];          // = Tm[tj, u0+2hi]
      a.y = frev[jb + 1];      // = Tm[tj, u0+2hi+1]
#pragma unroll
      for (int nt = 0; nt < 8; ++nt) {
        // B (4x16): lanes0-15 rows {u0,u0+1}, lanes16-31 rows {u0+2,u0+3}
        const int nn = (nt << 4) + l15;
        v2f bv;
        bv.x = Kt[cur][(u0 + (hi << 1)) * KT_P + nn];
        bv.y = Kt[cur][(u0 + (hi << 1) + 1) * KT_P + nn];
        acc[nt] = __builtin_amdgcn_wmma_f32_16x16x4_f32(
            false, a, false, bv, (short)0, acc[nt], false, false);
      }
    }
  }

  // D layout: VGPR i -> row M = i + 8*hi, col = lane&15
  const int trow0 = it * 128 + (w << 4) + (hi << 3);
#pragma unroll
  for (int nt = 0; nt < 8; ++nt)
#pragma unroll
    for (int i = 0; i < 8; ++i)
      cbuf[(size_t)(b * SEQL + trow0 + i) * N + (nt << 4) + l15] = acc[nt][i];
}

// ---------------------------------------------------------------------------
// Kernel 6: bandwidth-bound broadcast: ys[b,t,n,k] = C[n]*c[b,t,k] + D*f[b,t],
// plus c_fin = c[b, L-1, :].  256 MB streamed with non-temporal float4 stores.
// ---------------------------------------------------------------------------
__global__ __launch_bounds__(256) void k_bcast(const float* __restrict__ cbuf,
                                               const float* __restrict__ f,
                                               const float* __restrict__ Cv,
                                               const float* __restrict__ Dv,
                                               float* __restrict__ out) {
  const int bt = blockIdx.x;            // 0..4095
  const int b  = bt >> 11;
  const int t  = bt & (SEQL - 1);
  const int tid = threadIdx.x;
  __shared__ float row[N];
  __shared__ float cs[N];
  if (tid < N) {
    row[tid] = cbuf[(size_t)(b * SEQL + t) * N + tid];
    cs[tid]  = Cv[tid];
  }
  __syncthreads();
  const float dterm = Dv[0] * f[b * SEQL + t];
  float* ybase = out + 256 + (size_t)(b * SEQL + t) * N * N;
#pragma unroll
  for (int i = 0; i < 16; ++i) {
    int j  = tid + (i << 8);            // float4 index 0..4095 within (b,t)
    int n  = j >> 5;
    int k4 = (j & 31) << 2;
    float cn = cs[n];
    v4f v = {cn * row[k4]     + dterm, cn * row[k4 + 1] + dterm,
             cn * row[k4 + 2] + dterm, cn * row[k4 + 3] + dterm};
    __builtin_nontemporal_store(v, (v4f*)ybase + j);
  }
  if (t == SEQL - 1 && tid < N) out[b * N + tid] = row[tid];  // c_fin
}

// ---------------------------------------------------------------------------
extern "C" void kernel_launch(void* const* d_in, const int* in_sizes, int n_in,
                              void* d_out, int out_size, void* d_ws, size_t ws_size,
                              hipStream_t stream) {
  (void)in_sizes; (void)n_in; (void)out_size; (void)ws_size;
  const float* f  = (const float*)d_in[0];   // (2,2048,1)
  const float* A  = (const float*)d_in[1];   // (128,128)
  const float* Bv = (const float*)d_in[2];   // (128,1)
  const float* Cv = (const float*)d_in[3];   // (128,1)
  const float* Dv = (const float*)d_in[4];   // (1,)
  float* out = (float*)d_out;
  float* ws  = (float*)d_ws;

  // workspace layout (floats)
  float* Ad   = ws;                       // 16384
  float* Bd   = ws + 16384;               // 128
  float* M0   = ws + 16512;               // 16384
  float* M1   = ws + 32896;               // 16384
  float* Kext = ws + 49280;               // 2176*128 (128 zero rows + K[0..2047])
  float* cbuf = ws + 327808;              // 2*2048*128

  k_solve<<<dim3(1), dim3(128), 0, stream>>>(A, Bv, Ad, Bd);
  k_init<<<dim3(64), dim3(256), 0, stream>>>(Kext, Ad, Bd, M0);

  float* Mc = M0; float* Mn = M1;
  for (int k = 0; k < 11; ++k) {
    int len = 1 << k;
    k_expand<<<dim3((len * N + 255) / 256), dim3(256), 0, stream>>>(Kext, Mc, len);
    if (k < 10) {
      k_square<<<dim3(64), dim3(256), 0, stream>>>(Mc, Mn);
      float* tswap = Mc; Mc = Mn; Mn = tswap;
    }
  }

  k_conv<<<dim3(16, NBATCH), dim3(256), 0, stream>>>(f, Kext, cbuf);
  k_bcast<<<dim3(NBATCH * SEQL), dim3(256), 0, stream>>>(cbuf, f, Cv, Dv, out);
}